// Memory_75496935129331
// MI455X (gfx1250) — compile-verified
//
#include <hip/hip_runtime.h>
#include <hip/hip_bf16.h>

// Problem sizes (fixed by the reference).
#define M_ROWS 262144
#define K_DIM  256
#define V_DIM  256

typedef __attribute__((ext_vector_type(2))) float v2f;
typedef __attribute__((ext_vector_type(8))) float v8f;

__device__ __forceinline__ unsigned f32_sortable(float f) {
    unsigned u = __float_as_uint(f);
    return (u & 0x80000000u) ? ~u : (u | 0x80000000u);
}
__device__ __forceinline__ float f32_unsortable(unsigned u) {
    return (u & 0x80000000u) ? __uint_as_float(u ^ 0x80000000u)
                             : __uint_as_float(~u);
}

// ---------------------------------------------------------------------------
// Kernel 0: init the two sortable-uint max slots (everything else in ws is
// fully overwritten by its producer kernel every call).
// ---------------------------------------------------------------------------
__global__ void init_kernel(unsigned* __restrict__ umax) {
    if (threadIdx.x < 2) umax[threadIdx.x] = 0u;  // maps to -inf
}

// ---------------------------------------------------------------------------
// Kernel 1: scores.  s_w = (K @ key)/16, s_r = (K @ query)/16 via
// V_WMMA_F32_16X16X4_F32.  Each wave handles 16 rows; B columns 0/1 carry
// key/query, columns 2..15 read a zeroed LDS row (branchless, no EXEC churn).
// A layout (16x4 f32): lane n (0-15)/hi: a.x=A[M=n][K=2*hi], a.y=K=2*hi+1.
// B layout (4x16 f32): lane n: b.x=B[K=2*hi][N=n], b.y=B[K=2*hi+1][N=n].
// C layout: lanes(hi,n), vgpr v -> C[M=8*hi+v][N=n].
// ---------------------------------------------------------------------------
__global__ __launch_bounds__(256) void scores_kernel(
    const float* __restrict__ Kmat,
    const float* __restrict__ key,
    const float* __restrict__ query,
    float* __restrict__ Sw,
    float* __restrict__ Sr,
    unsigned* __restrict__ umax)
{
    // Rows: 0 = key, 1 = query, 2 = zeros (source for B columns 2..15).
    __shared__ float s_kq[3 * K_DIM];
    const int t = threadIdx.x;
    s_kq[t]             = key[t];
    s_kq[K_DIM + t]     = query[t];
    s_kq[2 * K_DIM + t] = 0.0f;
    __syncthreads();

    const int wave = t >> 5;
    const int lane = t & 31;
    const int n    = lane & 15;
    const int hi   = lane >> 4;
    const long m0  = (long)blockIdx.x * 128 + wave * 16;

    const float* rowp = Kmat + (m0 + n) * K_DIM + 2 * hi;
    const float* bsrc = s_kq + (n < 2 ? n : 2) * K_DIM + 2 * hi;

    v8f c = {};
    for (int k = 0; k < K_DIM; k += 4) {
        v2f a = *(const v2f*)(rowp + k);   // global_load_b64, streams K once
        v2f b = *(const v2f*)(bsrc + k);   // ds_load_b64, unconditional
        c = __builtin_amdgcn_wmma_f32_16x16x4_f32(
                false, a, false, b, (short)0, c, false, false);
    }

    // Columns 0/1 of C hold the two score vectors for rows m0+8*hi+v.
    float mloc = -3.0e38f;
    if (n < 2) {
        float* dst = (n == 0) ? Sw : Sr;
        #pragma unroll
        for (int v = 0; v < 8; ++v) {
            const long m = m0 + 8 * hi + v;
            const float s = c[v] * 0.0625f;   // 1/sqrt(256)
            dst[m] = s;
            mloc = fmaxf(mloc, s);
        }
    }
    float mw = (n == 0) ? mloc : -3.0e38f;
    float mr = (n == 1) ? mloc : -3.0e38f;
    #pragma unroll
    for (int off = 16; off > 0; off >>= 1) {
        mw = fmaxf(mw, __shfl_xor(mw, off, 32));
        mr = fmaxf(mr, __shfl_xor(mr, off, 32));
    }
    if (lane == 0) {
        atomicMax(&umax[0], f32_sortable(mw));
        atomicMax(&umax[1], f32_sortable(mr));
    }
}

// ---------------------------------------------------------------------------
// Kernel 2: exponentials + partial partition sums.  Overwrites Sw/Sr with
// e_w/e_r; per-block (fixed-order) partials -> zpart[b*3 + {0,1,2}].
// ---------------------------------------------------------------------------
__global__ __launch_bounds__(256) void sums_kernel(
    float* __restrict__ Sw, float* __restrict__ Sr,
    const unsigned* __restrict__ umax,
    float* __restrict__ zpart)
{
    const int t = threadIdx.x;
    const float mw = f32_unsortable(umax[0]);
    const float mr = f32_unsortable(umax[1]);

    float zw = 0.0f, zr = 0.0f, cc = 0.0f;
    #pragma unroll
    for (int j = 0; j < 4; ++j) {
        const long idx = (long)blockIdx.x * 1024 + j * 256 + t;
        const float ew = __expf(Sw[idx] - mw);
        const float er = __expf(Sr[idx] - mr);
        Sw[idx] = ew;
        Sr[idx] = er;
        zw += ew; zr += er; cc += ew * er;
    }
    #pragma unroll
    for (int off = 16; off > 0; off >>= 1) {
        zw += __shfl_xor(zw, off, 32);
        zr += __shfl_xor(zr, off, 32);
        cc += __shfl_xor(cc, off, 32);
    }
    __shared__ float red[8][3];
    const int wave = t >> 5;
    if ((t & 31) == 0) { red[wave][0] = zw; red[wave][1] = zr; red[wave][2] = cc; }
    __syncthreads();
    if (t == 0) {
        float a = 0, b = 0, d = 0;
        #pragma unroll
        for (int w = 0; w < 8; ++w) { a += red[w][0]; b += red[w][1]; d += red[w][2]; }
        zpart[blockIdx.x * 3 + 0] = a;
        zpart[blockIdx.x * 3 + 1] = b;
        zpart[blockIdx.x * 3 + 2] = d;
    }
}

// ---------------------------------------------------------------------------
// Kernel 3: weighted column sums via WMMA.  old_u = e_w @ V, r0_u = e_r @ V.
// 512 threads = 16 waves; wave w owns output columns [w*16, w*16+16).
// Each block covers 512 rows of V; partials -> vpart[b*512 + {col, 256+col}].
// A rows 0/1 carry e_w/e_r for 4 consecutive m's; rows 2..15 pull from a
// zeroed LDS row (branchless).
// ---------------------------------------------------------------------------
__global__ __launch_bounds__(512) void wsum_kernel(
    const float* __restrict__ Vmat,
    const float* __restrict__ Ew,
    const float* __restrict__ Er,
    float* __restrict__ vpart)
{
    // Rows: 0 = e_w, 1 = e_r, 2 = zeros (source for A rows 2..15).
    __shared__ float s_e[3 * 512];
    const int t  = threadIdx.x;
    const long mb = (long)blockIdx.x * 512;
    s_e[t]        = Ew[mb + t];
    s_e[512 + t]  = Er[mb + t];
    s_e[1024 + t] = 0.0f;
    __syncthreads();

    const int wave = t >> 5;
    const int lane = t & 31;
    const int n    = lane & 15;
    const int hi   = lane >> 4;
    const int c0   = wave * 16;

    const float* asrc = s_e + (n < 2 ? n : 2) * 512 + 2 * hi;
    const float* vp0  = Vmat + (mb + 2 * hi) * V_DIM + c0 + n;

    v8f c = {};
    for (int mm = 0; mm < 512; mm += 4) {
        v2f a = *(const v2f*)(asrc + mm);          // ds_load_b64, unconditional
        v2f b;
        b.x = vp0[(long)mm * V_DIM];               // V[mb+mm+2*hi][c0+n]
        b.y = vp0[(long)(mm + 1) * V_DIM];         // V[mb+mm+2*hi+1][c0+n]
        c = __builtin_amdgcn_wmma_f32_16x16x4_f32(
                false, a, false, b, (short)0, c, false, false);
    }
    // C row 0 (c[0], lanes hi==0) = old partial; row 1 (c[1]) = r0 partial.
    if (hi == 0) {
        vpart[(long)blockIdx.x * 512 + c0 + n]       = c[0];
        vpart[(long)blockIdx.x * 512 + 256 + c0 + n] = c[1];
    }
}

// ---------------------------------------------------------------------------
// Kernel 4: deterministic final reduction + output.
// ---------------------------------------------------------------------------
__global__ __launch_bounds__(256) void finalize_kernel(
    const float* __restrict__ value,
    const float* __restrict__ zpart,
    const float* __restrict__ vpart,
    float* __restrict__ out)
{
    const int t = threadIdx.x;
    __shared__ float sZ[3];
    if (t < 3) {
        float s = 0.0f;
        for (int i = 0; i < 256; ++i) s += zpart[i * 3 + t];
        sZ[t] = s;
    }
    __syncthreads();
    const float Zw = sZ[0], Zr = sZ[1], Cc = sZ[2];

    float oldu = 0.0f, r0u = 0.0f;
    for (int b = 0; b < 512; ++b) {
        oldu += vpart[(long)b * 512 + t];
        r0u  += vpart[(long)b * 512 + 256 + t];
    }
    const float oldv = oldu / Zw;
    const float coef = Cc / (Zw * Zr);
    out[t] = r0u / Zr + coef * (value[t] - oldv);
}

// ---------------------------------------------------------------------------
extern "C" void kernel_launch(void* const* d_in, const int* in_sizes, int n_in,
                              void* d_out, int out_size, void* d_ws, size_t ws_size,
                              hipStream_t stream)
{
    const float* key   = (const float*)d_in[0];
    const float* value = (const float*)d_in[1];
    const float* query = (const float*)d_in[2];
    const float* Kmat  = (const float*)d_in[3];
    const float* Vmat  = (const float*)d_in[4];
    float* out = (float*)d_out;

    float* ws = (float*)d_ws;
    float*    Sw    = ws;                                   // M floats
    float*    Sr    = ws + (size_t)M_ROWS;                  // M floats
    unsigned* umax  = (unsigned*)(ws + 2 * (size_t)M_ROWS); // 2 slots
    float*    zpart = ws + 2 * (size_t)M_ROWS + 8;          // 256*3 floats
    float*    vpart = ws + 2 * (size_t)M_ROWS + 1024;       // 512*512 floats

    init_kernel   <<<1,    32,  0, stream>>>(umax);
    scores_kernel <<<2048, 256, 0, stream>>>(Kmat, key, query, Sw, Sr, umax);
    sums_kernel   <<<256,  256, 0, stream>>>(Sw, Sr, umax, zpart);
    wsum_kernel   <<<512,  512, 0, stream>>>(Vmat, Sw, Sr, vpart);
    finalize_kernel<<<1,   256, 0, stream>>>(value, zpart, vpart, out);
}